// ProjectSolver_84078279786783
// MI455X (gfx1250) — compile-verified
//
#include <hip/hip_runtime.h>

#define GRID_N 30
#define IN_H_N 20
#define NCH    10
#define RPW    5                              // rows per wave (divides 30 AND 20 -> tiles are domain-pure)
#define WPB    8                              // wave32 waves per block
#define TILE_ELEMS (NCH * RPW * GRID_N)       // 1500 floats = 6000 B per wave

typedef unsigned int u32x4 __attribute__((ext_vector_type(4)));
typedef int          i32x4 __attribute__((ext_vector_type(4)));
typedef int          i32x8 __attribute__((ext_vector_type(8)));

// TDM descriptor for a 3D tile: dim0 = w (30 contiguous f32), dim1 = h (stride 30,
// take RPW), dim2 = c (stride 900, take 10). LDS layout = TDM linear fill order:
// idx = (c*RPW + h)*30 + w.
__device__ __forceinline__ void tdm_groups(unsigned long long ga, unsigned lds_addr,
                                           u32x4& g0, i32x8& g1, i32x4& g2) {
  g0.x = 1u;                                                  // count = 1 valid descriptor
  g0.y = lds_addr;                                            // D#.lds_addr (bytes)
  g0.z = (unsigned)(ga & 0xFFFFFFFFull);                      // global_addr[31:0]
  g0.w = (unsigned)((ga >> 32) & 0x1FFFFFFull) | (2u << 30);  // global_addr[56:32] | type=2
  g1[0] = 0x00020000;                // mask=0, data_size=2 (4 B), no pad/iterate/barrier
  g1[1] = GRID_N << 16;              // tensor_dim0 = 30            (bits 63:48)
  g1[2] = RPW    << 16;              // tensor_dim1 = 5             (bits 95:80)
  g1[3] = GRID_N << 16;              // tile_dim0   = 30            (bits 127:112)
  g1[4] = RPW | (NCH << 16);         // tile_dim1 = 5, tile_dim2 = 10
  g1[5] = GRID_N;                    // tensor_dim0_stride = 30     (bits 191:160)
  g1[6] = (GRID_N * GRID_N) << 16;   // tensor_dim1_stride = 900    (bits 223:208)
  g1[7] = 0;
  g2[0] = NCH;                       // tensor_dim2 = 10
  g2[1] = 0; g2[2] = 0; g2[3] = 0;   // no dim3, no iterate
}

__device__ __forceinline__ void tdm_load(u32x4 g0, i32x8 g1, i32x4 g2, i32x4 g3) {
#if __clang_major__ >= 23
  i32x8 z8 = {0, 0, 0, 0, 0, 0, 0, 0};
  __builtin_amdgcn_tensor_load_to_lds(g0, g1, g2, g3, z8, 0);
#else
  __builtin_amdgcn_tensor_load_to_lds(g0, g1, g2, g3, 0);
#endif
}

__device__ __forceinline__ void tdm_store(u32x4 g0, i32x8 g1, i32x4 g2, i32x4 g3) {
#if __clang_major__ >= 23
  i32x8 z8 = {0, 0, 0, 0, 0, 0, 0, 0};
  __builtin_amdgcn_tensor_store_from_lds(g0, g1, g2, g3, z8, 0);
#else
  __builtin_amdgcn_tensor_store_from_lds(g0, g1, g2, g3, 0);
#endif
}

// One wave32 per 5 rows; lane = column w. The 30-iteration flood fill collapses to
// "rightmost non-PASS to my left; fill iff it is SET" per row, computed with two
// ballots + firstbit-high. Tiles with h0 >= 20 are pure TDM copies.
__global__ __launch_bounds__(WPB * 32) void ProjectSolver_84078279786783_kernel(
    const float* __restrict__ x, float* __restrict__ out, int n_rows) {
  __shared__ float stage[WPB * TILE_ELEMS];   // 48 KB/block -> 6 blocks/WGP

  const int lane = (int)(threadIdx.x & 31u);
  const int wave = (int)(threadIdx.x >> 5u);
  const int row0 = ((int)blockIdx.x * WPB + wave) * RPW;
  if (row0 >= n_rows) return;

  const int b  = row0 / GRID_N;
  const int h0 = row0 - b * GRID_N;           // multiple of RPW -> tile stays in one batch
  const size_t base = (size_t)b * (NCH * GRID_N * GRID_N) + (size_t)h0 * GRID_N;

  float* myStage = &stage[wave * TILE_ELEMS];
  const unsigned lds_addr = (unsigned)(size_t)(void*)myStage;  // low 32 bits of generic = LDS offset

  u32x4 g0; i32x8 g1; i32x4 g2; i32x4 g3 = {0, 0, 0, 0};

  // ---- DMA input tile (10 c x 5 h x 30 w = 6 KB) into LDS ----
  tdm_groups((unsigned long long)(size_t)(const void*)(x + base), lds_addr, g0, g1, g2);
  tdm_load(g0, g1, g2, g3);
  __builtin_amdgcn_s_wait_tensorcnt(0);
  asm volatile("" ::: "memory");

  if (h0 < IN_H_N) {                          // in-domain tile (IN_H % RPW == 0)
    const bool active = lane < GRID_N;        // lanes 30,31 shadow lane 0, never write
    const int  w  = active ? lane : 0;
    const unsigned lt = (1u << lane) - 1u;    // lanemask_lt

#pragma unroll
    for (int i = 0; i < RPW; ++i) {
      float c[NCH];
#pragma unroll
      for (int k = 0; k < NCH; ++k)
        c[k] = myStage[(k * RPW + i) * GRID_N + w];

      float csum = 0.0f;
#pragma unroll
      for (int k = 1; k < NCH; ++k) csum += c[k];
      const bool hc  = csum > 0.5f;           // SET: cell has original color
      const bool isB = c[0] > 0.5f;           // background

      const unsigned setm = (unsigned)__ballot(hc);           // SET columns
      const unsigned npm  = (unsigned)__ballot(hc || !isB);   // non-PASS columns
      const unsigned np   = npm & lt;                         // non-PASS strictly left of me
      const int src = 31 - __builtin_clz(np | 1u);            // rightmost one (0 if np==0)
      const bool upd = isB && (np != 0u) && (((setm >> src) & 1u) != 0u);

      // Gathers must run with full EXEC (bpermute reads inactive lanes as 0).
      float f[NCH];
      f[0] = 0.0f;
#pragma unroll
      for (int k = 1; k < NCH; ++k) f[k] = __shfl(c[k], src, 32);

      // Only updated cells rewrite LDS; everyone else already holds the output.
      if (upd && active) {
#pragma unroll
        for (int k = 0; k < NCH; ++k)
          myStage[(k * RPW + i) * GRID_N + lane] = f[k];
      }
    }
  }

  // ---- DMA the (possibly updated) tile back out; DS must land before TDM reads ----
  asm volatile("" ::: "memory");
  asm volatile("s_wait_dscnt 0x0" ::: "memory");
  tdm_groups((unsigned long long)(size_t)(void*)(out + base), lds_addr, g0, g1, g2);
  tdm_store(g0, g1, g2, g3);
  // s_endpgm's implicit wait-idle drains TENSORcnt before LDS dealloc.
}

extern "C" void kernel_launch(void* const* d_in, const int* in_sizes, int n_in,
                              void* d_out, int out_size, void* d_ws, size_t ws_size,
                              hipStream_t stream) {
  (void)n_in; (void)out_size; (void)d_ws; (void)ws_size;
  const float* x  = (const float*)d_in[0];
  float* out      = (float*)d_out;
  const int total  = in_sizes[0];                  // B * 10 * 30 * 30
  const int n_rows = total / (NCH * GRID_N);       // B * 30 (multiple of RPW)
  const int waves  = (n_rows + RPW - 1) / RPW;
  const int blocks = (waves + WPB - 1) / WPB;
  ProjectSolver_84078279786783_kernel<<<blocks, WPB * 32, 0, stream>>>(x, out, n_rows);
}